// DistNStepTD_30073361007334
// MI455X (gfx1250) — compile-verified
//
#include <hip/hip_runtime.h>
#include <hip/hip_bf16.h>
#include <stdint.h>

#define NATOM 51
#define NACT  16
#define NSTEP 5
#define BLK   128

typedef __attribute__((ext_vector_type(2))) float v2f;
typedef __attribute__((ext_vector_type(8))) float v8f;

// One thread per batch element.
// Phase A: stream next_n_dist[b, next_n_action[b]] row (NT loads), scatter-
//          project into a private 51-bin LDS histogram (stride 51 = odd ->
//          conflict-free across lanes).
// Async:   meanwhile, DMA dist[b, action[b]] row into LDS via
//          global_load_async_to_lds_b32 (ASYNCcnt path, latency hiding).
// Phase B: td_err = -sum_j log(dist_row[j]) * proj[j]; deterministic block
//          reduction of weight*td_err.
__global__ __launch_bounds__(BLK) void c51_td_kernel(
    const float* __restrict__ dist,        // (B, 16, 51)
    const float* __restrict__ next_n_dist, // (B, 16, 51)
    const int*   __restrict__ action,      // (B,)
    const int*   __restrict__ next_n_action,
    const float* __restrict__ reward,      // (T, B)
    const unsigned char* __restrict__ done,
    const float* __restrict__ weight,
    const float* __restrict__ gamma_p,
    const float* __restrict__ vmin_p,
    const float* __restrict__ vmax_p,
    float* __restrict__ td_out,            // (B,)
    float* __restrict__ block_sums,        // (gridDim.x,)
    int Bsz)
{
    __shared__ float proj [BLK * NATOM];
    __shared__ float stage[BLK * NATOM];
    __shared__ float red  [BLK];

    const int tid = threadIdx.x;
    const int b   = blockIdx.x * BLK + tid;

    const float gamma = gamma_p[0];
    const float vmin  = vmin_p[0];
    const float vmax  = vmax_p[0];
    const float dz    = (vmax - vmin) / (float)(NATOM - 1);
    const float idz   = 1.0f / dz;

    const int act  = action[b];
    const int nact = next_n_action[b];
    const float* __restrict__ drow = dist        + ((size_t)b * NACT + act ) * NATOM;
    const float* __restrict__ nrow = next_n_dist + ((size_t)b * NACT + nact) * NATOM;

    // ---- kick off async DMA of the dist row into LDS (phase-B data) ----
    {
        uint32_t lds0 = (uint32_t)(uintptr_t)(&stage[tid * NATOM]);
        #pragma unroll
        for (int j = 0; j < NATOM; ++j) {
            uint32_t ldsa = lds0 + 4u * j;
            uint64_t ga   = (uint64_t)(uintptr_t)(drow + j);
            asm volatile("global_load_async_to_lds_b32 %0, %1, off"
                         :: "v"(ldsa), "v"(ga) : "memory");
        }
    }

    // ---- n-step discounted reward (coalesced across b, streamed once) ----
    float r = 0.0f, gp = 1.0f;
    #pragma unroll
    for (int t = 0; t < NSTEP; ++t) {
        r = fmaf(gp, __builtin_nontemporal_load(&reward[(size_t)t * Bsz + b]), r);
        gp *= gamma;
    }
    const float nd_gN = (done[b] ? 0.0f : 1.0f) * gp;   // not_done * gamma^T

    // ---- phase A: projection scatter into private LDS histogram ----
    float* __restrict__ myproj = &proj[tid * NATOM];
    #pragma unroll
    for (int j = 0; j < NATOM; ++j) myproj[j] = 0.0f;

    for (int j = 0; j < NATOM; ++j) {
        const float td  = __builtin_nontemporal_load(&nrow[j]);
        const float sup = fmaf(dz, (float)j, vmin);
        float tz = fmaf(nd_gN, sup, r);
        tz = fminf(vmax, fmaxf(vmin, tz));
        const float bb = (tz - vmin) * idz;
        float lf = floorf(bb);
        float uf = ceilf(bb);
        int   li = (int)lf;
        int   ui = (int)uf;
        if (li == ui) {                  // disappearing-mass fix (matches ref)
            if (ui > 0)            { li -= 1; lf -= 1.0f; }
            else /* li < NATOM-1 */{ ui += 1; uf += 1.0f; }
        }
        myproj[li] += td * (uf - bb);
        myproj[ui] += td * (bb - lf);
    }

    // ---- wait for async-staged dist row, then phase B ----
    asm volatile("s_wait_asynccnt 0x0" ::: "memory");

    const float* __restrict__ mystage = &stage[tid * NATOM];
    float acc = 0.0f;
    #pragma unroll
    for (int j = 0; j < NATOM; ++j)
        acc = fmaf(__logf(mystage[j]), myproj[j], acc);

    const float td_err = -acc;
    td_out[b] = td_err;

    // ---- deterministic block reduction of weight*td_err ----
    red[tid] = td_err * weight[b];
    __syncthreads();
    #pragma unroll
    for (int s = BLK / 2; s > 0; s >>= 1) {
        if (tid < s) red[tid] += red[tid + s];
        __syncthreads();
    }
    if (tid == 0) block_sums[blockIdx.x] = red[0];
}

// Final reduction of block partial sums via V_WMMA_F32_16X16X4_F32.
// A = all-ones (16x4)  =>  D[i][j] = sum_k B[k,j]. Summing D over all (i,j)
// entries of row 0 (and its row-8 twin) gives the exact chunk total regardless
// of the lane->(k,j) mapping, so each lane may load its v2f contiguously.
// n is guaranteed a multiple of 64 (B=65536, BLK=128 -> n=512): no guards,
// EXEC stays all-ones through the load+wmma clause.
__global__ __launch_bounds__(32) void c51_loss_reduce(
    const float* __restrict__ partials, float* __restrict__ loss_out,
    float invB, int nchunks)
{
    const int lane = threadIdx.x;       // wave32: one full wave, EXEC all-ones
    const v2f* __restrict__ p2 = (const v2f*)partials;

    v8f c = {};
    v2f a; a[0] = 1.0f; a[1] = 1.0f;

    for (int ch = 0; ch < nchunks; ++ch) {
        v2f bm = p2[ch * 32 + lane];    // single global_load_b64 per lane
        c = __builtin_amdgcn_wmma_f32_16x16x4_f32(
                /*neg_a=*/false, a, /*neg_b=*/false, bm,
                /*c_mod=*/(short)0, c, /*reuse_a=*/false, /*reuse_b=*/false);
    }

    // All D rows are identical (A == ones): lanes 0-15 of c[0] hold the 16
    // column sums (row 0), lanes 16-31 hold the same sums (row 8). Summing all
    // 32 lanes double-counts the total -> halve.
    float v = c[0];
    #pragma unroll
    for (int off = 16; off >= 1; off >>= 1)
        v += __shfl_xor(v, off, 32);

    if (lane == 0) loss_out[0] = v * 0.5f * invB;
}

extern "C" void kernel_launch(void* const* d_in, const int* in_sizes, int n_in,
                              void* d_out, int out_size, void* d_ws, size_t ws_size,
                              hipStream_t stream) {
    const float*         dist    = (const float*)d_in[0];
    const float*         nnd     = (const float*)d_in[1];
    const int*           act     = (const int*)d_in[2];
    const int*           nact    = (const int*)d_in[3];
    const float*         reward  = (const float*)d_in[4];
    const unsigned char* done    = (const unsigned char*)d_in[5]; // jax bool = 1 byte
    const float*         weight  = (const float*)d_in[6];
    const float*         gamma_p = (const float*)d_in[7];
    const float*         vmin_p  = (const float*)d_in[8];
    const float*         vmax_p  = (const float*)d_in[9];

    const int B  = in_sizes[2];          // action count == batch (65536)
    const int nb = B / BLK;              // 512 blocks -> 512 partials (8 x 64)

    float* out      = (float*)d_out;     // out[0]=loss, out[1..B]=td_err
    float* partials = (float*)d_ws;      // nb floats of scratch

    c51_td_kernel<<<nb, BLK, 0, stream>>>(dist, nnd, act, nact, reward, done,
                                          weight, gamma_p, vmin_p, vmax_p,
                                          out + 1, partials, B);
    c51_loss_reduce<<<1, 32, 0, stream>>>(partials, out, 1.0f / (float)B,
                                          nb / 64);
}